// EngramCell_82935818485852
// MI455X (gfx1250) — compile-verified
//
#include <hip/hip_runtime.h>
#include <hip/hip_bf16.h>

// EngramCell for MI455X (gfx1250, wave32).
// f32 end-to-end via V_WMMA_F32_16X16X4_F32 (bandwidth-bound: ~80MB streamed
// vs 1.27 GFLOP -> 16 FLOP/B; fp32 WMMA keeps precision while the weight
// stream sets the time). Skinny GEMMs (B=32): one wave32 per 16x16 output
// tile; split-K + f32 atomics for the K=3072 GEMMs. The two big weight
// streams use CDNA5 async movement: TDM tensor_load_to_lds for W_int (two 12KB
// descriptors per 24KB split-K slab, TENSORcnt-tracked) and
// global_load_async_to_lds_b128 for W_out (ASYNCcnt-tracked, double-buffered).

#define Bb 32
#define Dd 1024
#define Hh 1024
#define Mm 2048
#define H3 3072

typedef __attribute__((ext_vector_type(2))) float v2f;
typedef __attribute__((ext_vector_type(8))) float v8f;
typedef __attribute__((ext_vector_type(4))) unsigned int u32x4;
typedef __attribute__((ext_vector_type(8))) int i32x8;
typedef __attribute__((ext_vector_type(4))) int i32x4;

// ---------------------------------------------------------------- WMMA core
__device__ __forceinline__ v8f wmma4(v2f a, v2f b, v8f c) {
  // (neg_a, A, neg_b, B, c_mod, C, reuse_a, reuse_b)
  return __builtin_amdgcn_wmma_f32_16x16x4_f32(false, a, false, b, (short)0, c,
                                               false, false);
}

// A fragment from row-major P (rows = tile dim), also serves B^T when the
// "B" matrix is stored row-major [N,K] (e.g. eff_mem for z @ eff_mem^T).
// ISA layout: lanes 0-15 -> K={k,k+1}, lanes 16-31 -> K={k+2,k+3}.
__device__ __forceinline__ v2f frag_rows(const float* __restrict__ P, int ld,
                                         int r0, int k) {
  int lane = threadIdx.x & 31;
  const float* p = P + (size_t)(r0 + (lane & 15)) * ld + k + ((lane >> 4) << 1);
  return *(const v2f*)p;  // 8B-aligned -> global_load_b64
}

__device__ __forceinline__ v2f frag_rows_relu(const float* __restrict__ P,
                                              int ld, int r0, int k) {
  v2f f = frag_rows(P, ld, r0, k);
  f.x = fmaxf(f.x, 0.0f);
  f.y = fmaxf(f.y, 0.0f);
  return f;
}

// B fragment from row-major B[K,N] (strided along K), also serves A^T access.
__device__ __forceinline__ v2f frag_strided(const float* __restrict__ P, int ld,
                                            int c0, int k) {
  int lane = threadIdx.x & 31;
  int cc = c0 + (lane & 15);
  int kk = k + ((lane >> 4) << 1);
  v2f f;
  f.x = P[(size_t)kk * ld + cc];
  f.y = P[(size_t)(kk + 1) * ld + cc];
  return f;
}

// B fragment from an LDS tile laid out row-major [rows][16].
__device__ __forceinline__ v2f frag_lds(const float* lp, int k) {
  int lane = threadIdx.x & 31;
  int cc = lane & 15;
  int kk = k + ((lane >> 4) << 1);
  v2f f;
  f.x = lp[kk * 16 + cc];
  f.y = lp[(kk + 1) * 16 + cc];
  return f;
}

// Generic LDS pointer -> 32-bit LDS byte offset (low 32 bits of the shared
// aperture address are the allocation-relative LDS address).
__device__ __forceinline__ uint32_t lds_offset(const void* p) {
  return (uint32_t)(uintptr_t)p;
}

// C/D layout: VGPR v holds rows m0+v (lanes 0-15) and m0+v+8 (lanes 16-31).
#define TILE_EPILOGUE(rb, col)              \
  int lane_ep = threadIdx.x & 31;           \
  int col = n0 + (lane_ep & 15);            \
  int rb = m0 + ((lane_ep >> 4) << 3);

// ------------------------------------------- CDNA5 async/TDM feature gates
#if __has_builtin(__builtin_amdgcn_tensor_load_to_lds)
#define HAVE_TDM 1
#if __has_include(<hip/amd_detail/amd_gfx1250_TDM.h>)
#define TDM_6ARG 1  // amdgpu-toolchain / therock headers -> 6-arg builtin
#endif
#endif

#if __has_builtin(__builtin_amdgcn_s_wait_tensorcnt)
#define WAIT_TENSOR(n) __builtin_amdgcn_s_wait_tensorcnt(n)
#else
#define WAIT_TENSOR(n) asm volatile("s_wait_tensorcnt " #n ::: "memory")
#endif

#if __has_builtin(__builtin_amdgcn_s_wait_asynccnt)
#define WAIT_ASYNC(n) __builtin_amdgcn_s_wait_asynccnt(n)
#else
#define WAIT_ASYNC(n) asm volatile("s_wait_asynccnt " #n ::: "memory")
#endif

// Async global->LDS copy, 16B per lane (VDST = LDS byte address VGPR).
__device__ __forceinline__ void async_load_b128(uint32_t lds_byte,
                                                const float* g) {
  asm volatile("global_load_async_to_lds_b128 %0, %1, off"
               :: "v"(lds_byte), "v"(g) : "memory");
}

#ifdef HAVE_TDM
// Issue one TDM descriptor: 2-D tile, 16 f32 wide x `rows` tall, row stride
// `stride_elems` f32, from `gsrc` into LDS at byte offset `lds_byte`.
// D# layout per CDNA5 ISA 8.3/8.4 (count=1, data_size=4B, type=2).
__device__ __forceinline__ void tdm_load_tile16(uint32_t lds_byte,
                                                const float* gsrc, int rows,
                                                int stride_elems) {
  uint64_t ga = (uint64_t)(uintptr_t)gsrc;
  u32x4 g0;
  g0[0] = 1u;                                            // count=1, user mode
  g0[1] = lds_byte;                                      // lds_addr
  g0[2] = (uint32_t)ga;                                  // global_addr[31:0]
  g0[3] = (uint32_t)((ga >> 32) & 0x1FFFFFFu) | (2u << 30);  // addr[56:32]|type
  i32x8 g1;
  g1[0] = (int)(2u << 16);               // workgroup_mask=0, data_size=4B
  g1[1] = (int)(16u << 16);              // tensor_dim0[15:0]=16 (bits 63:48)
  g1[2] = (int)(((uint32_t)rows & 0xFFFFu) << 16);  // dim0 hi=0, dim1 lo
  g1[3] = (int)((((uint32_t)rows >> 16) & 0xFFFFu) | (16u << 16));  // tile_dim0
  g1[4] = (int)((uint32_t)rows & 0xFFFFu);          // tile_dim1, tile_dim2=0
  g1[5] = (int)(uint32_t)stride_elems;              // tensor_dim0_stride lo32
  g1[6] = 0;                                        // stride hi, dim1_stride
  g1[7] = 0;
  i32x4 z4 = {0, 0, 0, 0};
#ifdef TDM_6ARG
  i32x8 z8 = {0, 0, 0, 0, 0, 0, 0, 0};
  __builtin_amdgcn_tensor_load_to_lds(g0, g1, z4, z4, z8, 0);
#else
  __builtin_amdgcn_tensor_load_to_lds(g0, g1, z4, z4, 0);
#endif
}
#endif

// ------------------------------------------------------------- reductions
__device__ __forceinline__ float block_sum(float v, float* sdata) {
  int lane = threadIdx.x & 31, wid = threadIdx.x >> 5;
  #pragma unroll
  for (int off = 16; off > 0; off >>= 1) v += __shfl_down(v, off, 32);
  if (lane == 0) sdata[wid] = v;
  __syncthreads();
  int nw = blockDim.x >> 5;
  v = (threadIdx.x < nw) ? sdata[threadIdx.x] : 0.0f;
  if (wid == 0) {
    #pragma unroll
    for (int off = 4; off > 0; off >>= 1) v += __shfl_down(v, off, 32);
    if (lane == 0) sdata[0] = v;
  }
  __syncthreads();
  float r = sdata[0];
  __syncthreads();
  return r;
}

__device__ __forceinline__ float block_max(float v, float* sdata) {
  int lane = threadIdx.x & 31, wid = threadIdx.x >> 5;
  #pragma unroll
  for (int off = 16; off > 0; off >>= 1) v = fmaxf(v, __shfl_down(v, off, 32));
  if (lane == 0) sdata[wid] = v;
  __syncthreads();
  int nw = blockDim.x >> 5;
  v = (threadIdx.x < nw) ? sdata[threadIdx.x] : -3.4e38f;
  if (wid == 0) {
    #pragma unroll
    for (int off = 4; off > 0; off >>= 1) v = fmaxf(v, __shfl_down(v, off, 32));
    if (lane == 0) sdata[0] = v;
  }
  __syncthreads();
  float r = sdata[0];
  __syncthreads();
  return r;
}

// -------------------------------------------------- K0: eff_mem + row rnorm
__global__ void k_effmem(const float* __restrict__ bank,
                         const float* __restrict__ trace,
                         float* __restrict__ em, float* __restrict__ minv) {
  __shared__ float sdata[9];
  int row = blockIdx.x;
  const float* bp = bank + (size_t)row * Hh;
  const float* tp = trace + (size_t)row * Hh;
  float* ep = em + (size_t)row * Hh;
  float ss = 0.0f;
  #pragma unroll
  for (int i = 0; i < Hh / 256; ++i) {
    int c = threadIdx.x + i * 256;
    float v = bp[c] + 0.5f * tp[c];
    ep[c] = v;
    ss += v * v;
  }
  ss = block_sum(ss, sdata);
  if (threadIdx.x == 0) minv[row] = rsqrtf(fmaxf(ss, 1e-12f));
}

// -------------------------------------- K1: z_raw = relu(x @ W_enc + b_enc)
__global__ void k_enc(const float* __restrict__ x, const float* __restrict__ W,
                      const float* __restrict__ bias,
                      float* __restrict__ combined) {
  const int NT = Hh / 16;
  int m0 = (blockIdx.x / NT) << 4;
  int n0 = (blockIdx.x % NT) << 4;
  v8f c = {};
  for (int k = 0; k < Dd; k += 4) {
    v2f a = frag_rows(x, Dd, m0, k);
    v2f b = frag_strided(W, Hh, n0, k);
    c = wmma4(a, b, c);
  }
  TILE_EPILOGUE(rb, col);
  float bv = bias[col];
  #pragma unroll
  for (int v = 0; v < 8; ++v)
    combined[(size_t)(rb + v) * H3 + col] = fmaxf(c[v] + bv, 0.0f);
}

// ------------------- K2/K9: layernorm (+ optional pre-relu, optional 1/||.||)
__global__ void k_layernorm(const float* __restrict__ src, int lds_,
                            float* __restrict__ dst, int ldd,
                            const float* __restrict__ gamma,
                            const float* __restrict__ beta,
                            float* __restrict__ invnrm, int relu) {
  __shared__ float sdata[9];
  int row = blockIdx.x;
  const float* sp = src + (size_t)row * lds_;
  float* dp = dst + (size_t)row * ldd;
  float v[Hh / 256];
  float s = 0.0f;
  #pragma unroll
  for (int i = 0; i < Hh / 256; ++i) {
    float t = sp[threadIdx.x + i * 256];
    if (relu) t = fmaxf(t, 0.0f);
    v[i] = t;
    s += t;
  }
  s = block_sum(s, sdata);
  float mu = s * (1.0f / (float)Hh);
  float q = 0.0f;
  #pragma unroll
  for (int i = 0; i < Hh / 256; ++i) {
    float d = v[i] - mu;
    q += d * d;
  }
  q = block_sum(q, sdata);
  float rstd = rsqrtf(q * (1.0f / (float)Hh) + 1e-6f);
  float ss = 0.0f;
  #pragma unroll
  for (int i = 0; i < Hh / 256; ++i) {
    int cc = threadIdx.x + i * 256;
    float o = (v[i] - mu) * rstd * gamma[cc] + beta[cc];
    dp[cc] = o;
    ss += o * o;
  }
  if (invnrm != nullptr) {
    ss = block_sum(ss, sdata);
    if (threadIdx.x == 0) invnrm[row] = rsqrtf(fmaxf(ss, 1e-12f));
  }
}

// --------------------------------------------- K3: prev_h -> combined[:,2H:]
__global__ void k_copy_prevh(const float* __restrict__ prev,
                             float* __restrict__ combined) {
  int idx = blockIdx.x * blockDim.x + threadIdx.x;
  if (idx < Bb * Hh) {
    int b = idx >> 10, h = idx & (Hh - 1);
    combined[(size_t)b * H3 + 2 * Hh + h] = prev[idx];
  }
}

// ---------------------- K4: logits = (z @ em^T) * zinv[b] * minv[m] / 0.75
__global__ void k_sim(const float* __restrict__ combinedZ,
                      const float* __restrict__ em,
                      const float* __restrict__ zinv,
                      const float* __restrict__ minv,
                      float* __restrict__ attn) {
  const int NT = Mm / 16;
  int m0 = (blockIdx.x / NT) << 4;
  int n0 = (blockIdx.x % NT) << 4;
  v8f c = {};
  for (int k = 0; k < Hh; k += 4) {
    v2f a = frag_rows(combinedZ, H3, m0, k);
    v2f b = frag_rows(em, Hh, n0, k);  // em row-major [M,H] == B^T access
    c = wmma4(a, b, c);
  }
  TILE_EPILOGUE(rb, col);
  float sm = minv[col] * (4.0f / 3.0f);  // fold 1/0.75 temperature
  #pragma unroll
  for (int v = 0; v < 8; ++v) {
    int r = rb + v;
    attn[(size_t)r * Mm + col] = c[v] * zinv[r] * sm;
  }
}

// ------------------------------------------------------------- K5: softmax
__global__ void k_softmax(float* __restrict__ attn) {
  __shared__ float sdata[9];
  float* p = attn + (size_t)blockIdx.x * Mm;
  float vals[Mm / 256];
  float mx = -3.4e38f;
  #pragma unroll
  for (int i = 0; i < Mm / 256; ++i) {
    vals[i] = p[threadIdx.x + i * 256];
    mx = fmaxf(mx, vals[i]);
  }
  mx = block_max(mx, sdata);
  float s = 0.0f;
  #pragma unroll
  for (int i = 0; i < Mm / 256; ++i) {
    vals[i] = __expf(vals[i] - mx);
    s += vals[i];
  }
  s = block_sum(s, sdata);
  float inv = 1.0f / s;
  #pragma unroll
  for (int i = 0; i < Mm / 256; ++i) p[threadIdx.x + i * 256] = vals[i] * inv;
}

// ---------------------------------------- K6: m_t = attn @ em -> combined[:,H:]
__global__ void k_mt(const float* __restrict__ attn, const float* __restrict__ em,
                     float* __restrict__ combined) {
  const int NT = Hh / 16;
  int m0 = (blockIdx.x / NT) << 4;
  int n0 = (blockIdx.x % NT) << 4;
  v8f c = {};
  for (int k = 0; k < Mm; k += 4) {
    v2f a = frag_rows(attn, Mm, m0, k);
    v2f b = frag_strided(em, Hh, n0, k);
    c = wmma4(a, b, c);
  }
  TILE_EPILOGUE(rb, col);
  #pragma unroll
  for (int v = 0; v < 8; ++v)
    combined[(size_t)(rb + v) * H3 + Hh + col] = c[v];
}

// ------------------------------------------------ K7: dst[b][j] = bias[j]
__global__ void k_bias_init(float* __restrict__ dst,
                            const float* __restrict__ bias, int n) {
  int idx = blockIdx.x * blockDim.x + threadIdx.x;
  if (idx < Bb * n) dst[idx] = bias[idx % n];
}

// ------------- K8: hidden += combined @ W_int (split-K=8, f32 atomics).
// Weight slab (16 cols x 384 rows, 24KB) is DMA'd by the Tensor Data Mover
// into LDS, double-buffered on TENSORcnt; falls back to direct global loads.
__global__ void k_int(const float* __restrict__ Amat, const float* __restrict__ W,
                      float* __restrict__ out) {
  const int NT = H3 / 16;           // 192
  const int TILES = (Bb / 16) * NT; // 384
  int tile = blockIdx.x % TILES;
  int split = blockIdx.x / TILES;
  int m0 = (tile / NT) << 4;
  int n0 = (tile % NT) << 4;
  const int KC = H3 / 8;            // 384
  int ks = split * KC;
  v8f c = {};
#ifdef HAVE_TDM
  __shared__ float wtile[2][(KC / 2) * 16];  // 2 x 12KB
  const float* gbase = W + (size_t)ks * H3 + n0;
  tdm_load_tile16(lds_offset(&wtile[0][0]), gbase, KC / 2, H3);
  tdm_load_tile16(lds_offset(&wtile[1][0]), gbase + (size_t)(KC / 2) * H3,
                  KC / 2, H3);
  WAIT_TENSOR(1);  // first slab resident
  for (int k = 0; k < KC / 2; k += 4) {
    v2f a = frag_rows(Amat, H3, m0, ks + k);
    c = wmma4(a, frag_lds(&wtile[0][0], k), c);
  }
  WAIT_TENSOR(0);  // second slab resident
  for (int k = 0; k < KC / 2; k += 4) {
    v2f a = frag_rows(Amat, H3, m0, ks + KC / 2 + k);
    c = wmma4(a, frag_lds(&wtile[1][0], k), c);
  }
#else
  for (int k = ks; k < ks + KC; k += 4) {
    int kp = k + 32;
    if (kp < H3) __builtin_prefetch(&W[(size_t)kp * H3 + n0], 0, 0);
    v2f a = frag_rows(Amat, H3, m0, k);
    v2f b = frag_strided(W, H3, n0, k);
    c = wmma4(a, b, c);
  }
#endif
  TILE_EPILOGUE(rb, col);
  #pragma unroll
  for (int v = 0; v < 8; ++v)
    atomicAdd(&out[(size_t)(rb + v) * H3 + col], c[v]);
}

// --------- K10: hpre += relu(hidden) @ W_out (split-K=8, relu fused in A).
// W stream staged through LDS with global_load_async_to_lds_b128 (16B/lane,
// 8 rows per issue), double-buffered on ASYNCcnt.
__global__ void k_out(const float* __restrict__ hidden,
                      const float* __restrict__ W, float* __restrict__ out) {
  __shared__ float wtile[2][32 * 16];  // 2 x 2KB chunks (32 K-rows x 16 cols)
  const int NT = Hh / 16;              // 64
  const int TILES = (Bb / 16) * NT;    // 128
  int tile = blockIdx.x % TILES;
  int split = blockIdx.x / TILES;
  int m0 = (tile / NT) << 4;
  int n0 = (tile % NT) << 4;
  const int KC = H3 / 8;               // 384 -> 12 chunks of 32
  int ks = split * KC;
  int lane = threadIdx.x & 31;
  int srow = lane >> 2;                // 0..7
  int scol = (lane & 3) << 2;          // 0,4,8,12 (floats)
  const float* gbase = W + (size_t)ks * Hh + n0;
  uint32_t l0 = lds_offset(&wtile[0][0]);
  uint32_t l1 = lds_offset(&wtile[1][0]);
  #pragma unroll
  for (int j = 0; j < 4; ++j)  // preload chunk 0
    async_load_b128(l0 + (uint32_t)(((j * 8 + srow) * 16 + scol) * 4),
                    gbase + (size_t)(j * 8 + srow) * Hh + scol);
  v8f c = {};
  for (int ch = 0; ch < KC / 32; ++ch) {
    int buf = ch & 1;
    if (ch + 1 < KC / 32) {
      const float* gnext = gbase + (size_t)(ch + 1) * 32 * Hh;
      uint32_t lb = buf ? l0 : l1;
      #pragma unroll
      for (int j = 0; j < 4; ++j)
        async_load_b128(lb + (uint32_t)(((j * 8 + srow) * 16 + scol) * 4),
                        gnext + (size_t)(j * 8 + srow) * Hh + scol);
      WAIT_ASYNC(4);  // current chunk's 4 transfers complete (in-order)
    } else {
      WAIT_ASYNC(0);
    }
    const float* lp = &wtile[buf][0];
    #pragma unroll
    for (int kk = 0; kk < 32; kk += 4) {
      v2f a = frag_rows_relu(hidden, H3, m0, ks + ch * 32 + kk);
      c = wmma4(a, frag_lds(lp, kk), c);
    }
  }
  TILE_EPILOGUE(rb, col);
  #pragma unroll
  for (int v = 0; v < 8; ++v)
    atomicAdd(&out[(size_t)(rb + v) * Hh + col], c[v]);
}

// -------- K11: new_trace = clip(0.95*trace + (0.05/32)*(attn^T @ z), +-0.1)
// (jax threefry noise term, magnitude 5e-5 post-lr, is not reproducible here)
__global__ void k_trace(const float* __restrict__ attn,
                        const float* __restrict__ combinedZ,
                        const float* __restrict__ trace,
                        float* __restrict__ outTrace) {
  const int NT = Hh / 16;
  int m0 = (blockIdx.x / NT) << 4;  // over M
  int n0 = (blockIdx.x % NT) << 4;  // over H
  v8f c = {};
  #pragma unroll
  for (int k = 0; k < Bb; k += 4) {
    v2f a = frag_strided(attn, Mm, m0, k);       // A[m][k] = attn[k][m]
    v2f b = frag_strided(combinedZ, H3, n0, k);  // B[k][n] = z_t[k][n]
    c = wmma4(a, b, c);
  }
  const float s = 0.05f / 32.0f;
  TILE_EPILOGUE(rb, col);
  #pragma unroll
  for (int v = 0; v < 8; ++v) {
    size_t idx = (size_t)(rb + v) * Hh + col;
    float t = trace[idx] * 0.95f + c[v] * s;
    outTrace[idx] = fminf(fmaxf(t, -0.1f), 0.1f);
  }
}

// ------------------------------------------------------------------ launch
extern "C" void kernel_launch(void* const* d_in, const int* in_sizes, int n_in,
                              void* d_out, int out_size, void* d_ws,
                              size_t ws_size, hipStream_t stream) {
  const float* x      = (const float*)d_in[0];
  const float* prev_h = (const float*)d_in[1];
  const float* trace  = (const float*)d_in[2];
  const float* bank   = (const float*)d_in[3];
  const float* W_enc  = (const float*)d_in[4];
  const float* b_enc  = (const float*)d_in[5];
  const float* gamma1 = (const float*)d_in[6];
  const float* beta1  = (const float*)d_in[7];
  const float* W_int  = (const float*)d_in[8];
  const float* b_int  = (const float*)d_in[9];
  const float* W_out  = (const float*)d_in[10];
  const float* b_out  = (const float*)d_in[11];
  const float* gamma2 = (const float*)d_in[12];
  const float* beta2  = (const float*)d_in[13];

  float* ws = (float*)d_ws;
  float* em       = ws;                                  // M*H
  float* minv     = em + (size_t)Mm * Hh;                // M
  float* attn     = minv + Mm;                           // B*M
  float* combined = attn + (size_t)Bb * Mm;              // B*3H
  float* hidden   = combined + (size_t)Bb * H3;          // B*3H
  float* hpre     = hidden + (size_t)Bb * H3;            // B*H
  float* zinv     = hpre + (size_t)Bb * Hh;              // B

  float* out_h  = (float*)d_out;
  float* out_tr = out_h + (size_t)Bb * Hh;

  k_effmem<<<Mm, 256, 0, stream>>>(bank, trace, em, minv);
  k_enc<<<(Bb / 16) * (Hh / 16), 32, 0, stream>>>(x, W_enc, b_enc, combined);
  k_layernorm<<<Bb, 256, 0, stream>>>(combined, H3, combined, H3, gamma1,
                                      beta1, zinv, 0);
  k_copy_prevh<<<(Bb * Hh + 255) / 256, 256, 0, stream>>>(prev_h, combined);
  k_sim<<<(Bb / 16) * (Mm / 16), 32, 0, stream>>>(combined, em, zinv, minv,
                                                  attn);
  k_softmax<<<Bb, 256, 0, stream>>>(attn);
  k_mt<<<(Bb / 16) * (Hh / 16), 32, 0, stream>>>(attn, em, combined);
  k_bias_init<<<(Bb * H3 + 255) / 256, 256, 0, stream>>>(hidden, b_int, H3);
  k_int<<<(Bb / 16) * (H3 / 16) * 8, 32, 0, stream>>>(combined, W_int, hidden);
  k_bias_init<<<(Bb * Hh + 255) / 256, 256, 0, stream>>>(hpre, b_out, Hh);
  k_out<<<(Bb / 16) * (Hh / 16) * 8, 32, 0, stream>>>(hidden, W_out, hpre);
  k_layernorm<<<Bb, 256, 0, stream>>>(hpre, Hh, out_h, Hh, gamma2, beta2,
                                      nullptr, 1);
  k_trace<<<(Mm / 16) * (Hh / 16), 32, 0, stream>>>(attn, combined, trace,
                                                    out_tr);
}